// DyFilterTopoNet_8323646619655
// MI455X (gfx1250) — compile-verified
//
#include <hip/hip_runtime.h>
#include <math.h>

typedef float v2f __attribute__((ext_vector_type(2)));
typedef float v8f __attribute__((ext_vector_type(8)));

#define BB 8
#define FF 64
#define HH 80
#define WW 80
#define HWP (HH*WW)
#define NLIN 4624
#define NLIN2 2312
#define KSPLIT 16

static inline int cdiv(int a, int b) { return (a + b - 1) / b; }

// ---------------- helpers ----------------
__device__ inline float bilerp(const float* __restrict__ p, int Hin, int Win,
                               float sy, float sx) {
  int y0 = (int)floorf(sy), x0 = (int)floorf(sx);
  float fy = sy - (float)y0, fx = sx - (float)x0;
  int y1 = y0 + 1, x1 = x0 + 1;
  y0 = min(max(y0, 0), Hin - 1); y1 = min(max(y1, 0), Hin - 1);
  x0 = min(max(x0, 0), Win - 1); x1 = min(max(x1, 0), Win - 1);
  float v00 = p[y0 * Win + x0], v01 = p[y0 * Win + x1];
  float v10 = p[y1 * Win + x0], v11 = p[y1 * Win + x1];
  return v00 * (1.f - fy) * (1.f - fx) + v01 * (1.f - fy) * fx +
         v10 * fy * (1.f - fx) + v11 * fy * fx;
}

// ---------------- stage kernels ----------------
__global__ void k_maxf(const float* __restrict__ x, float* __restrict__ x1) {
  int idx = blockIdx.x * blockDim.x + threadIdx.x;
  if (idx >= BB * HWP) return;
  int b = idx / HWP, p = idx % HWP;
  float m = x[(size_t)b * FF * HWP + p];
  for (int f = 1; f < FF; ++f)
    m = fmaxf(m, x[((size_t)b * FF + f) * HWP + p]);
  x1[idx] = m;
}

__global__ void k_avg3_tanh(const float* __restrict__ in, float* __restrict__ out,
                            int N, int Hin, int Win) {
  int Ho = Hin - 2, Wo = Win - 2;
  int idx = blockIdx.x * blockDim.x + threadIdx.x;
  if (idx >= N * Ho * Wo) return;
  int n = idx / (Ho * Wo), r = idx % (Ho * Wo), i = r / Wo, j = r % Wo;
  const float* p = in + (size_t)n * Hin * Win + i * Win + j;
  float s = 0.f;
  for (int dy = 0; dy < 3; ++dy)
    for (int dx = 0; dx < 3; ++dx) s += p[dy * Win + dx];
  out[idx] = tanhf(s * (1.f / 9.f));
}

__global__ void k_max3(const float* __restrict__ in, float* __restrict__ out,
                       int N, int Hin, int Win) {
  int Ho = Hin - 2, Wo = Win - 2;
  int idx = blockIdx.x * blockDim.x + threadIdx.x;
  if (idx >= N * Ho * Wo) return;
  int n = idx / (Ho * Wo), r = idx % (Ho * Wo), i = r / Wo, j = r % Wo;
  const float* p = in + (size_t)n * Hin * Win + i * Win + j;
  float m = -INFINITY;
  for (int dy = 0; dy < 3; ++dy)
    for (int dx = 0; dx < 3; ++dx) m = fmaxf(m, p[dy * Win + dx]);
  out[idx] = m;
}

__global__ void k_resize(const float* __restrict__ in, float* __restrict__ out,
                         int N, int Hin, int Win, int Ho, int Wo) {
  int idx = blockIdx.x * blockDim.x + threadIdx.x;
  if (idx >= N * Ho * Wo) return;
  int n = idx / (Ho * Wo), r = idx % (Ho * Wo), i = r / Wo, j = r % Wo;
  float sy = (i + 0.5f) * (float)Hin / (float)Ho - 0.5f;
  float sx = (j + 0.5f) * (float)Win / (float)Wo - 0.5f;
  out[idx] = bilerp(in + (size_t)n * Hin * Win, Hin, Win, sy, sx);
}

// conv 2x2 pad1 -> 81x81 then BN+tanh
__global__ void k_conv2_bn_tanh(const float* __restrict__ in, float* __restrict__ out,
                                const float* __restrict__ w, const float* __restrict__ g,
                                const float* __restrict__ bb, int N) {
  int idx = blockIdx.x * blockDim.x + threadIdx.x;
  if (idx >= N * 81 * 81) return;
  int n = idx / (81 * 81), r = idx % (81 * 81), i = r / 81, j = r % 81;
  const float* p = in + (size_t)n * HWP;
  float s = 0.f;
  for (int di = 0; di < 2; ++di) {
    int yy = i - 1 + di;
    if (yy < 0 || yy >= HH) continue;
    for (int dj = 0; dj < 2; ++dj) {
      int xx = j - 1 + dj;
      if (xx < 0 || xx >= WW) continue;
      s += p[yy * WW + xx] * w[di * 2 + dj];
    }
  }
  float gs = g[0] * rsqrtf(1.f + 1e-5f);
  out[idx] = tanhf(s * gs + bb[0]);
}

// conv 4x4 pad1 -> 79x79 then BN+tanh
__global__ void k_conv4_bn_tanh(const float* __restrict__ in, float* __restrict__ out,
                                const float* __restrict__ w, const float* __restrict__ g,
                                const float* __restrict__ bb, int N) {
  int idx = blockIdx.x * blockDim.x + threadIdx.x;
  if (idx >= N * 79 * 79) return;
  int n = idx / (79 * 79), r = idx % (79 * 79), i = r / 79, j = r % 79;
  const float* p = in + (size_t)n * HWP;
  float s = 0.f;
  for (int di = 0; di < 4; ++di) {
    int yy = i - 1 + di;
    if (yy < 0 || yy >= HH) continue;
    for (int dj = 0; dj < 4; ++dj) {
      int xx = j - 1 + dj;
      if (xx < 0 || xx >= WW) continue;
      s += p[yy * WW + xx] * w[di * 4 + dj];
    }
  }
  float gs = g[0] * rsqrtf(1.f + 1e-5f);
  out[idx] = tanhf(s * gs + bb[0]);
}

__global__ void k_msb_combine(const float* __restrict__ t81, const float* __restrict__ t79,
                              float* __restrict__ out, const float* __restrict__ wd,
                              const float* __restrict__ gd, const float* __restrict__ bd,
                              int N) {
  int idx = blockIdx.x * blockDim.x + threadIdx.x;
  if (idx >= N * HWP) return;
  int n = idx / HWP, r = idx % HWP, i = r / WW, j = r % WW;
  float r1 = bilerp(t81 + (size_t)n * 81 * 81, 81, 81,
                    (i + 0.5f) * 81.f / 80.f - 0.5f, (j + 0.5f) * 81.f / 80.f - 0.5f);
  float r2 = bilerp(t79 + (size_t)n * 79 * 79, 79, 79,
                    (i + 0.5f) * 79.f / 80.f - 0.5f, (j + 0.5f) * 79.f / 80.f - 0.5f);
  float gs = gd[0] * rsqrtf(1.f + 1e-5f);
  out[idx] = (r1 * wd[0] + r2 * wd[1]) * gs + bd[0];
}

__global__ void k_meanF(const float* __restrict__ chan, float* __restrict__ mf) {
  int idx = blockIdx.x * blockDim.x + threadIdx.x;
  if (idx >= BB * HWP) return;
  int b = idx / HWP, p = idx % HWP;
  float s = 0.f;
  for (int f = 0; f < FF; ++f) s += chan[((size_t)b * FF + f) * HWP + p];
  mf[idx] = s * (1.f / FF);
}

// t[b,f,k1,k2] = mean(cw_w)*tileAvg(chan) + mean(cw_b) + tileAvg(meanF)
__global__ void k_tileT(const float* __restrict__ chan, const float* __restrict__ mf,
                        float* __restrict__ t, const float* __restrict__ cww,
                        const float* __restrict__ cwb) {
  int idx = blockIdx.x * blockDim.x + threadIdx.x;
  if (idx >= BB * FF * 25) return;
  int b = idx / (FF * 25), r = idx % (FF * 25), f = r / 25, kk = r % 25;
  int k1 = kk / 5, k2 = kk % 5;
  const float* cp = chan + ((size_t)b * FF + f) * HWP;
  const float* mp = mf + (size_t)b * HWP;
  float s = 0.f, sm = 0.f;
  for (int m = 0; m < 16; ++m) {
    int row = (k1 * 16 + m) * WW + k2 * 16;
    for (int n = 0; n < 16; ++n) { s += cp[row + n]; sm += mp[row + n]; }
  }
  float cwm = 0.f, cbm = 0.f;
  for (int c = 0; c < 16; ++c) { cwm += cww[c]; cbm += cwb[c]; }
  cwm *= (1.f / 16.f); cbm *= (1.f / 16.f);
  t[idx] = cwm * (s * (1.f / 256.f)) + cbm + sm * (1.f / 256.f);
}

__global__ void k_chan_matvec(const float* __restrict__ t, const float* __restrict__ dw,
                              const float* __restrict__ db, float* __restrict__ cw) {
  int idx = blockIdx.x * blockDim.x + threadIdx.x;
  if (idx >= BB * FF * 25) return;
  int b = idx / (FF * 25), r = idx % (FF * 25), g = r / 25, kk = r % 25;
  float s = db[g];
  for (int f = 0; f < FF; ++f)
    s += dw[g * FF + f] * t[((size_t)b * FF + f) * 25 + kk];
  cw[idx] = s;
}

__global__ void k_spat_softmax(const float* __restrict__ sp, const float* __restrict__ sfw,
                               const float* __restrict__ sfb, float* __restrict__ out) {
  int idx = blockIdx.x * blockDim.x + threadIdx.x;
  if (idx >= BB * HWP) return;
  int b = idx / HWP, p = idx % HWP;
  float v = sp[idx];
  float lg[25], mx = -INFINITY;
  for (int c = 0; c < 25; ++c) {
    lg[c] = v * (sfw[c] + 1.f) + sfb[c];
    mx = fmaxf(mx, lg[c]);
  }
  float sum = 0.f;
  for (int c = 0; c < 25; ++c) { lg[c] = expf(lg[c] - mx); sum += lg[c]; }
  float inv = 1.f / sum;
  for (int c = 0; c < 25; ++c)
    out[((size_t)b * 25 + c) * HWP + p] = lg[c] * inv;
}

__global__ void k_sumF(const float* __restrict__ xe, float* __restrict__ S) {
  int idx = blockIdx.x * blockDim.x + threadIdx.x;
  if (idx >= BB * HWP) return;
  int b = idx / HWP, p = idx % HWP;
  float s = 0.f;
  for (int f = 0; f < FF; ++f) s += xe[((size_t)b * FF + f) * HWP + p];
  S[idx] = s;
}

__global__ void k_apply(const float* __restrict__ xe, const float* __restrict__ S,
                        const float* __restrict__ cw, const float* __restrict__ sw,
                        float* __restrict__ y2) {
  int idx = blockIdx.x * blockDim.x + threadIdx.x;
  if (idx >= BB * HWP) return;
  int b = idx / HWP, p = idx % HWP, pi = p / WW, pj = p % WW;
  float acc_s = 0.f;
  for (int i = 0; i < 5; ++i) {
    int yy = pi + i - 2;
    if (yy < 0 || yy >= HH) continue;
    for (int j = 0; j < 5; ++j) {
      int xx = pj + j - 2;
      if (xx < 0 || xx >= WW) continue;
      acc_s += S[(size_t)b * HWP + yy * WW + xx] *
               sw[((size_t)b * 25 + i * 5 + j) * HWP + p];
    }
  }
  float acc_c = 0.f;
  for (int f = 0; f < FF; ++f) {
    const float* xp = xe + ((size_t)b * FF + f) * HWP;
    const float* cp = cw + ((size_t)b * FF + f) * 25;
    for (int i = 0; i < 5; ++i) {
      int yy = pi + i - 2;
      if (yy < 0 || yy >= HH) continue;
      for (int j = 0; j < 5; ++j) {
        int xx = pj + j - 2;
        if (xx < 0 || xx >= WW) continue;
        acc_c += xp[yy * WW + xx] * cp[i * 5 + j];
      }
    }
  }
  y2[(size_t)b * 2 * HWP + p] = acc_s;
  y2[(size_t)b * 2 * HWP + HWP + p] = acc_c;
}

// conv 5x5 (2->6) valid + relu + maxpool2 : (B,2,80,80)->(B,6,38,38)
__global__ void k_conv1pool(const float* __restrict__ in, const float* __restrict__ w,
                            const float* __restrict__ bias, float* __restrict__ out) {
  int idx = blockIdx.x * blockDim.x + threadIdx.x;
  if (idx >= BB * 6 * 38 * 38) return;
  int b = idx / (6 * 38 * 38), r = idx % (6 * 38 * 38);
  int oc = r / (38 * 38); r %= 38 * 38;
  int py = r / 38, px = r % 38;
  float mx = -INFINITY;
  for (int dy = 0; dy < 2; ++dy)
    for (int dx = 0; dx < 2; ++dx) {
      int cy = 2 * py + dy, cx = 2 * px + dx;
      float s = bias[oc];
      for (int ic = 0; ic < 2; ++ic) {
        const float* ip = in + ((size_t)(b * 2 + ic)) * HWP + cy * WW + cx;
        const float* wp = w + (oc * 2 + ic) * 25;
        for (int ky = 0; ky < 5; ++ky)
          for (int kx = 0; kx < 5; ++kx) s += ip[ky * WW + kx] * wp[ky * 5 + kx];
      }
      mx = fmaxf(mx, fmaxf(s, 0.f));
    }
  out[idx] = mx;
}

// conv 5x5 (6->16) valid + relu + maxpool2 + flatten to (16,4624); rows>=8 zero
__global__ void k_conv2pool_flat(const float* __restrict__ in, const float* __restrict__ w,
                                 const float* __restrict__ bias, float* __restrict__ feat) {
  int idx = blockIdx.x * blockDim.x + threadIdx.x;
  if (idx >= 16 * NLIN) return;
  int b = idx / NLIN, r = idx % NLIN;
  if (b >= BB) { feat[idx] = 0.f; return; }
  int oc = r / 289, rr = r % 289, py = rr / 17, px = rr % 17;
  float mx = -INFINITY;
  for (int dy = 0; dy < 2; ++dy)
    for (int dx = 0; dx < 2; ++dx) {
      int cy = 2 * py + dy, cx = 2 * px + dx;
      float s = bias[oc];
      for (int ic = 0; ic < 6; ++ic) {
        const float* ip = in + ((size_t)(b * 6 + ic)) * 38 * 38 + cy * 38 + cx;
        const float* wp = w + (oc * 6 + ic) * 25;
        for (int ky = 0; ky < 5; ++ky)
          for (int kx = 0; kx < 5; ++kx) s += ip[ky * 38 + kx] * wp[ky * 5 + kx];
      }
      mx = fmaxf(mx, fmaxf(s, 0.f));
    }
  feat[idx] = mx;
}

#define WMMA4(aa, bb, cc) \
  __builtin_amdgcn_wmma_f32_16x16x4_f32(false, (aa), false, (bb), (short)0, (cc), false, false)

// split-K WMMA GEMM: part[chunk][16][N] = A(16,K) x W(N,K)^T over this K-slice.
// Branch-free N-edge: out-of-range lanes read clamped row N-1 (valid memory);
// their garbage output columns are simply not stored. Valid columns are exact
// since every D element is an independent dot product.
__global__ __launch_bounds__(32) void k_gemm_wmma(const float* __restrict__ A,
                                                  const float* __restrict__ Wm,
                                                  float* __restrict__ part,
                                                  int N, int K, int kchunk) {
  int tile = blockIdx.x, chunk = blockIdx.y;
  int lane = threadIdx.x & 31;
  int half = lane >> 4, l = lane & 15;
  int n0 = tile * 16;
  int k0 = chunk * kchunk;
  int kend = min(K, k0 + kchunk);
  int n = n0 + l;
  int nc = min(n, N - 1);
  const float* arow = A + (size_t)l * K + 2 * half;   // A-frag: lane(l,half) -> (M=l, K=k+2h..)
  const float* wrow = Wm + (size_t)nc * K + 2 * half; // B-frag: lane(l,half) -> (N=n, K=k+2h..)
  v8f acc = {0.f, 0.f, 0.f, 0.f, 0.f, 0.f, 0.f, 0.f};
  int k = k0;
  for (; k + 16 <= kend; k += 16) {  // 4 WMMAs / iter, clustered b64 loads
    v2f a0 = *(const v2f*)(arow + k);
    v2f a1 = *(const v2f*)(arow + k + 4);
    v2f a2 = *(const v2f*)(arow + k + 8);
    v2f a3 = *(const v2f*)(arow + k + 12);
    v2f b0 = *(const v2f*)(wrow + k);
    v2f b1 = *(const v2f*)(wrow + k + 4);
    v2f b2 = *(const v2f*)(wrow + k + 8);
    v2f b3 = *(const v2f*)(wrow + k + 12);
    __builtin_prefetch(wrow + k + 512, 0, 1);  // stream W ~2KB ahead
    acc = WMMA4(a0, b0, acc);
    acc = WMMA4(a1, b1, acc);
    acc = WMMA4(a2, b2, acc);
    acc = WMMA4(a3, b3, acc);
  }
  for (; k < kend; k += 4) {  // remainder (kchunk is 16-aligned, normally unused)
    v2f a = *(const v2f*)(arow + k);
    v2f b = *(const v2f*)(wrow + k);
    acc = WMMA4(a, b, acc);
  }
  if (n < N) {
#pragma unroll
    for (int r = 0; r < 8; ++r) {
      int m = r + 8 * half;  // C/D layout: vgpr r, lanes16-31 -> M=r+8
      part[((size_t)chunk * 16 + m) * N + n] = acc[r];
    }
  }
}

__global__ void k_reduce_bias_relu(const float* __restrict__ part,
                                   const float* __restrict__ bias,
                                   float* __restrict__ out, int N, int nchunks) {
  int idx = blockIdx.x * blockDim.x + threadIdx.x;
  if (idx >= 16 * N) return;
  int m = idx / N, n = idx % N;
  if (m >= BB) { out[idx] = 0.f; return; }
  float s = bias[n];
  for (int c = 0; c < nchunks; ++c) s += part[((size_t)c * 16 + m) * N + n];
  out[idx] = fmaxf(s, 0.f);
}

__global__ void k_l3(const float* __restrict__ h2, const float* __restrict__ w,
                     const float* __restrict__ bias, float* __restrict__ out) {
  __shared__ float sm[256];
  int b = blockIdx.x, t = threadIdx.x;
  float s = 0.f;
  for (int k = t; k < NLIN2; k += 256) s += h2[(size_t)b * NLIN2 + k] * w[k];
  sm[t] = s;
  __syncthreads();
  for (int st = 128; st > 0; st >>= 1) {
    if (t < st) sm[t] += sm[t + st];
    __syncthreads();
  }
  if (t == 0) out[b] = 1.f / (1.f + expf(-(sm[0] + bias[0])));
}

// ---------------- launcher ----------------
extern "C" void kernel_launch(void* const* d_in, const int* in_sizes, int n_in,
                              void* d_out, int out_size, void* d_ws, size_t ws_size,
                              hipStream_t stream) {
  (void)in_sizes; (void)n_in; (void)out_size; (void)ws_size;
  const float* x = (const float*)d_in[0];
  const float* mc_w1 = (const float*)d_in[1];  const float* mc_g1 = (const float*)d_in[2];
  const float* mc_b1 = (const float*)d_in[3];  const float* mc_w2 = (const float*)d_in[4];
  const float* mc_g2 = (const float*)d_in[5];  const float* mc_b2 = (const float*)d_in[6];
  const float* mc_wd = (const float*)d_in[7];  const float* mc_gd = (const float*)d_in[8];
  const float* mc_bd = (const float*)d_in[9];
  const float* ms_w1 = (const float*)d_in[10]; const float* ms_g1 = (const float*)d_in[11];
  const float* ms_b1 = (const float*)d_in[12]; const float* ms_w2 = (const float*)d_in[13];
  const float* ms_g2 = (const float*)d_in[14]; const float* ms_b2 = (const float*)d_in[15];
  const float* ms_wd = (const float*)d_in[16]; const float* ms_gd = (const float*)d_in[17];
  const float* ms_bd = (const float*)d_in[18];
  const float* sf_w = (const float*)d_in[19];  const float* sf_b = (const float*)d_in[20];
  const float* cf_cw_w = (const float*)d_in[21]; const float* cf_cw_b = (const float*)d_in[22];
  const float* cf_dw = (const float*)d_in[23]; const float* cf_db = (const float*)d_in[24];
  const float* dp_c1w = (const float*)d_in[25]; const float* dp_c1b = (const float*)d_in[26];
  const float* dp_c2w = (const float*)d_in[27]; const float* dp_c2b = (const float*)d_in[28];
  const float* dp_l1w = (const float*)d_in[29]; const float* dp_l1b = (const float*)d_in[30];
  const float* dp_l2w = (const float*)d_in[31]; const float* dp_l2b = (const float*)d_in[32];
  const float* dp_l3w = (const float*)d_in[33]; const float* dp_l3b = (const float*)d_in[34];
  const float* dr_c1w = (const float*)d_in[35]; const float* dr_c2w = (const float*)d_in[36];
  const float* dr_l1w = (const float*)d_in[37]; const float* dr_l2w = (const float*)d_in[38];
  const float* dr_l3w = (const float*)d_in[39];
  float* out = (float*)d_out;

  float* ws = (float*)d_ws;
  size_t off = 0;
  auto alloc = [&](size_t n) { float* p = ws + off; off += n; return p; };
  float* x_enc  = alloc((size_t)BB * FF * HWP);
  float* x1_enc = alloc((size_t)BB * HWP);
  float* tmpA   = alloc((size_t)BB * FF * 81 * 81);
  float* tmpB   = alloc((size_t)BB * FF * 81 * 81);
  float* chan   = alloc((size_t)BB * FF * HWP);
  float* spat   = alloc((size_t)BB * HWP);
  float* x1bufA = alloc((size_t)BB * 81 * 81);
  float* x1bufB = alloc((size_t)BB * 81 * 81);
  float* mf     = alloc((size_t)BB * HWP);
  float* tileT  = alloc((size_t)BB * FF * 25);
  float* chanW  = alloc((size_t)BB * FF * 25);
  float* spatW  = alloc((size_t)BB * 25 * HWP);
  float* S      = alloc((size_t)BB * HWP);
  float* y2     = alloc((size_t)BB * 2 * HWP);
  float* c1out  = alloc((size_t)BB * 6 * 38 * 38);
  float* feat   = alloc((size_t)16 * NLIN);
  float* h1     = alloc((size_t)16 * NLIN);
  float* h2     = alloc((size_t)16 * NLIN2);
  float* part   = alloc((size_t)KSPLIT * 16 * NLIN);

  const int TB = 256;
  const int NP = BB * FF;  // 512 planes

  // x1 = max over F
  k_maxf<<<cdiv(BB * HWP, TB), TB, 0, stream>>>(x, x1bufA);

  // encoder(x): 3x (avg3+tanh, max3), then resize 68->80
  k_avg3_tanh<<<cdiv(NP * 78 * 78, TB), TB, 0, stream>>>(x, tmpA, NP, 80, 80);
  k_max3<<<cdiv(NP * 76 * 76, TB), TB, 0, stream>>>(tmpA, tmpB, NP, 78, 78);
  k_avg3_tanh<<<cdiv(NP * 74 * 74, TB), TB, 0, stream>>>(tmpB, tmpA, NP, 76, 76);
  k_max3<<<cdiv(NP * 72 * 72, TB), TB, 0, stream>>>(tmpA, tmpB, NP, 74, 74);
  k_avg3_tanh<<<cdiv(NP * 70 * 70, TB), TB, 0, stream>>>(tmpB, tmpA, NP, 72, 72);
  k_max3<<<cdiv(NP * 68 * 68, TB), TB, 0, stream>>>(tmpA, tmpB, NP, 70, 70);
  k_resize<<<cdiv(NP * HWP, TB), TB, 0, stream>>>(tmpB, x_enc, NP, 68, 68, 80, 80);

  // encoder(x1)
  k_avg3_tanh<<<cdiv(BB * 78 * 78, TB), TB, 0, stream>>>(x1bufA, x1bufB, BB, 80, 80);
  k_max3<<<cdiv(BB * 76 * 76, TB), TB, 0, stream>>>(x1bufB, x1bufA, BB, 78, 78);
  k_avg3_tanh<<<cdiv(BB * 74 * 74, TB), TB, 0, stream>>>(x1bufA, x1bufB, BB, 76, 76);
  k_max3<<<cdiv(BB * 72 * 72, TB), TB, 0, stream>>>(x1bufB, x1bufA, BB, 74, 74);
  k_avg3_tanh<<<cdiv(BB * 70 * 70, TB), TB, 0, stream>>>(x1bufA, x1bufB, BB, 72, 72);
  k_max3<<<cdiv(BB * 68 * 68, TB), TB, 0, stream>>>(x1bufB, x1bufA, BB, 70, 70);
  k_resize<<<cdiv(BB * HWP, TB), TB, 0, stream>>>(x1bufA, x1_enc, BB, 68, 68, 80, 80);

  // MSB (chan stream, 512 planes)
  k_conv2_bn_tanh<<<cdiv(NP * 81 * 81, TB), TB, 0, stream>>>(x_enc, tmpA, mc_w1, mc_g1, mc_b1, NP);
  k_conv4_bn_tanh<<<cdiv(NP * 79 * 79, TB), TB, 0, stream>>>(x_enc, tmpB, mc_w2, mc_g2, mc_b2, NP);
  k_msb_combine<<<cdiv(NP * HWP, TB), TB, 0, stream>>>(tmpA, tmpB, chan, mc_wd, mc_gd, mc_bd, NP);

  // MSB (spat stream, 8 planes)
  k_conv2_bn_tanh<<<cdiv(BB * 81 * 81, TB), TB, 0, stream>>>(x1_enc, x1bufA, ms_w1, ms_g1, ms_b1, BB);
  k_conv4_bn_tanh<<<cdiv(BB * 79 * 79, TB), TB, 0, stream>>>(x1_enc, x1bufB, ms_w2, ms_g2, ms_b2, BB);
  k_msb_combine<<<cdiv(BB * HWP, TB), TB, 0, stream>>>(x1bufA, x1bufB, spat, ms_wd, ms_gd, ms_bd, BB);

  // dynamic filters
  k_meanF<<<cdiv(BB * HWP, TB), TB, 0, stream>>>(chan, mf);
  k_tileT<<<cdiv(BB * FF * 25, TB), TB, 0, stream>>>(chan, mf, tileT, cf_cw_w, cf_cw_b);
  k_chan_matvec<<<cdiv(BB * FF * 25, TB), TB, 0, stream>>>(tileT, cf_dw, cf_db, chanW);
  k_spat_softmax<<<cdiv(BB * HWP, TB), TB, 0, stream>>>(spat, sf_w, sf_b, spatW);
  k_sumF<<<cdiv(BB * HWP, TB), TB, 0, stream>>>(x_enc, S);
  k_apply<<<cdiv(BB * HWP, TB), TB, 0, stream>>>(x_enc, S, chanW, spatW, y2);

  // decoders (WMMA GEMMs with split-K). kchunk multiple of 16 so the unrolled
  // 4-WMMA loop covers every chunk exactly (4624 = 15*304 + 64).
  const int kchunk = cdiv(NLIN, 16 * KSPLIT) * 16;  // 304
  auto run_decoder = [&](const float* c1w, const float* c1b, const float* c2w,
                         const float* c2b, const float* l1w, const float* l1b,
                         const float* l2w, const float* l2b, const float* l3w,
                         const float* l3b, float* outp) {
    k_conv1pool<<<cdiv(BB * 6 * 38 * 38, TB), TB, 0, stream>>>(y2, c1w, c1b, c1out);
    k_conv2pool_flat<<<cdiv(16 * NLIN, TB), TB, 0, stream>>>(c1out, c2w, c2b, feat);
    {
      dim3 g(cdiv(NLIN, 16), KSPLIT);
      k_gemm_wmma<<<g, 32, 0, stream>>>(feat, l1w, part, NLIN, NLIN, kchunk);
    }
    k_reduce_bias_relu<<<cdiv(16 * NLIN, TB), TB, 0, stream>>>(part, l1b, h1, NLIN, KSPLIT);
    {
      dim3 g(cdiv(NLIN2, 16), KSPLIT);
      k_gemm_wmma<<<g, 32, 0, stream>>>(h1, l2w, part, NLIN2, NLIN, kchunk);
    }
    k_reduce_bias_relu<<<cdiv(16 * NLIN2, TB), TB, 0, stream>>>(part, l2b, h2, NLIN2, KSPLIT);
    k_l3<<<BB, 256, 0, stream>>>(h2, l3w, l3b, outp);
  };

  run_decoder(dp_c1w, dp_c1b, dp_c2w, dp_c2b, dp_l1w, dp_l1b, dp_l2w, dp_l2b,
              dp_l3w, dp_l3b, out);          // pre -> out[0..7]
  run_decoder(dr_c1w, dp_c1b, dr_c2w, dp_c2b, dr_l1w, dp_l1b, dr_l2w, dp_l2b,
              dr_l3w, dp_l3b, out + BB);     // rec -> out[8..15]
}